// AdaptiveEmbeddingT2I_80126909874336
// MI455X (gfx1250) — compile-verified
//
#include <hip/hip_runtime.h>
#include <math.h>

// Problem constants (match reference)
#define Bsz   48
#define Tlen  50
#define Dd    1024
#define Rr    36
#define SMOOTHF 10.0f

typedef float v2f __attribute__((ext_vector_type(2)));
typedef float v8f __attribute__((ext_vector_type(8)));

// ---------------------------------------------------------------------------
// Kernel 1: BatchNorm1d training stats per channel d over (B, R).
// img_embed layout: (B, R, D). Consecutive threads -> consecutive d (coalesced).
// ---------------------------------------------------------------------------
__global__ __launch_bounds__(256)
void bn_stats_kernel(const float* __restrict__ img,
                     float* __restrict__ mean_out,
                     float* __restrict__ rsig_out) {
    int d = blockIdx.x * blockDim.x + threadIdx.x;
    if (d >= Dd) return;
    float s = 0.0f, ss = 0.0f;
    for (int b = 0; b < Bsz; ++b) {
        for (int r = 0; r < Rr; ++r) {
            float v = img[(b * Rr + r) * Dd + d];
            s += v;
            ss = fmaf(v, v, ss);
        }
    }
    const float n = (float)(Bsz * Rr);
    float m   = s / n;
    float var = ss / n - m * m;            // biased variance (jnp.var default)
    mean_out[d] = m;
    rsig_out[d] = rsqrtf(var + 1e-5f);
}

// ---------------------------------------------------------------------------
// Kernel 2: ragged masked mean pooling over first lens[b] words, plus fused
// L2-normalized copy (cap_vec). One block per caption b, 256 threads over D.
// ---------------------------------------------------------------------------
__global__ __launch_bounds__(256)
void cap_pool_kernel(const float* __restrict__ cap,
                     const int*   __restrict__ lens,
                     float* __restrict__ cap_repr,
                     float* __restrict__ capn) {
    const int b   = blockIdx.x;
    const int tid = threadIdx.x;
    const int len = lens[b];
    const float inv = 1.0f / (float)len;

    float vals[Dd / 256];
    float ssq = 0.0f;
    #pragma unroll
    for (int j = 0; j < Dd / 256; ++j) {
        int d = tid + j * 256;
        float s = 0.0f;
        for (int t = 0; t < len; ++t)
            s += cap[(b * Tlen + t) * Dd + d];
        float r = s * inv;
        vals[j] = r;
        cap_repr[b * Dd + d] = r;
        ssq = fmaf(r, r, ssq);
    }
    // block reduction of sum of squares (wave32 shuffles, then cross-wave LDS)
    __shared__ float wsum[8];
    for (int off = 16; off > 0; off >>= 1)
        ssq += __shfl_down(ssq, off, 32);
    int lane = tid & 31, wid = tid >> 5;
    if (lane == 0) wsum[wid] = ssq;
    __syncthreads();
    if (tid == 0) {
        float tot = 0.0f;
        #pragma unroll
        for (int w = 0; w < 8; ++w) tot += wsum[w];
        wsum[0] = 1.0f / (sqrtf(tot) + 1e-8f);
    }
    __syncthreads();
    const float innorm = wsum[0];
    #pragma unroll
    for (int j = 0; j < Dd / 256; ++j) {
        int d = tid + j * 256;
        capn[b * Dd + d] = vals[j] * innorm;
    }
}

// ---------------------------------------------------------------------------
// Kernel 3: FiLM GEMMs via V_WMMA_F32_16X16X4_F32 (fp32-exact).
//   O[m,n] = sum_k A[m,k] * W[n,k] + bias[n]        (A = cap_repr, 48x1024)
// One wave (32 threads) per 16x16 output tile; K-loop of 256 WMMAs.
// Lane layout per CDNA5 ISA 7.12.2:
//   A 16x4 fp32 : lanes 0-15 M=lane, VGPR0=K0/VGPR1=K1; lanes 16-31 K2/K3
//   C/D 16x16   : VGPRj -> M = j + 8*(lane>=16), N = lane&15
// ---------------------------------------------------------------------------
__global__ __launch_bounds__(32)
void film_gemm_kernel(const float* __restrict__ A,
                      const float* __restrict__ Wg, const float* __restrict__ bg,
                      const float* __restrict__ Wb, const float* __restrict__ bb,
                      float* __restrict__ Og,       float* __restrict__ Ob) {
    const int n0    = blockIdx.x * 16;
    const int m0    = blockIdx.y * 16;
    const int which = blockIdx.z;
    const float* W    = which ? Wb : Wg;
    const float* bias = which ? bb : bg;
    float*       O    = which ? Ob : Og;

    const int lane  = threadIdx.x;
    const int half  = lane >> 4;       // 0 or 1
    const int l15   = lane & 15;
    const int kbase = half * 2;        // K offset for this lane half

    const float* arow = A + (m0 + l15) * Dd;   // A row (M = l15)
    const float* brow = W + (n0 + l15) * Dd;   // W row n holds B column n

    v8f acc = {0.f, 0.f, 0.f, 0.f, 0.f, 0.f, 0.f, 0.f};
    for (int k0 = 0; k0 < Dd; k0 += 4) {
        v2f a, b;
        a.x = arow[k0 + kbase];
        a.y = arow[k0 + kbase + 1];
        b.x = brow[k0 + kbase];
        b.y = brow[k0 + kbase + 1];
        acc = __builtin_amdgcn_wmma_f32_16x16x4_f32(
                  /*neg_a=*/false, a, /*neg_b=*/false, b,
                  /*c_mod=*/(short)0, acc, /*reuse_a=*/false, /*reuse_b=*/false);
    }
    const float bv = bias[n0 + l15];
    #pragma unroll
    for (int j = 0; j < 8; ++j) {
        int m = m0 + j + half * 8;
        O[m * Dd + (n0 + l15)] = acc[j] + bv;
    }
}

// ---------------------------------------------------------------------------
// Kernel 4: fused FiLM-apply + fovea softmax + region mean + L2 norm + cosine.
// Block = (image i, tile of 8 captions). 256 threads over D (4 channels each).
// Each thread keeps the 36 normalized region values for its channel in
// registers and reuses them across 8 captions -> adapted tensor (340 MB)
// never hits memory. exp throughput bound.
// ---------------------------------------------------------------------------
__global__ __launch_bounds__(256)
void fovea_sims_kernel(const float* __restrict__ img,    // (B,R,D)
                       const float* __restrict__ mean,
                       const float* __restrict__ rsig,
                       const float* __restrict__ gammas, // (B,D)
                       const float* __restrict__ betas,  // (B,D)
                       const float* __restrict__ capn,   // (B,D) L2-normalized
                       float* __restrict__ out) {        // (B,B)
    const int i   = blockIdx.x;
    const int c0  = blockIdx.y * 8;
    const int tid = threadIdx.x;

    float dotv[8] = {0.f, 0.f, 0.f, 0.f, 0.f, 0.f, 0.f, 0.f};
    float nrmv[8] = {0.f, 0.f, 0.f, 0.f, 0.f, 0.f, 0.f, 0.f};

    for (int j = 0; j < Dd / 256; ++j) {
        const int d  = tid + j * 256;
        const float mu = mean[d];
        const float rs = rsig[d];
        if (j + 1 < Dd / 256)   // hint next channel chunk into cache
            __builtin_prefetch(&img[(i * Rr) * Dd + d + 256], 0, 3);

        // normalized region values for this channel, kept in VGPRs
        float x[Rr];
        #pragma unroll
        for (int r = 0; r < Rr; ++r)
            x[r] = (img[(i * Rr + r) * Dd + d] - mu) * rs;

        #pragma unroll
        for (int cc = 0; cc < 8; ++cc) {
            const int c = c0 + cc;
            const float g  = 1.0f + gammas[c * Dd + d];
            const float bt = betas[c * Dd + d];
            // pass 1: max over regions (recompute fma, saves 36 VGPRs)
            float mx = -3.0e38f;
            #pragma unroll
            for (int r = 0; r < Rr; ++r)
                mx = fmaxf(mx, fmaf(x[r], g, bt));
            // pass 2: softmax numerator sums
            float se = 0.0f, sea = 0.0f;
            #pragma unroll
            for (int r = 0; r < Rr; ++r) {
                float ar = fmaf(x[r], g, bt);
                float e  = __expf((ar - mx) * SMOOTHF);
                se += e;
                sea = fmaf(e, ar, sea);
            }
            const float v = sea / (se * (float)Rr);   // (softmax*adapted).mean(R)
            dotv[cc] = fmaf(v, capn[c * Dd + d], dotv[cc]);
            nrmv[cc] = fmaf(v, v, nrmv[cc]);
        }
    }

    // reduce dot & ||img_vec||^2 over the 1024 channels (8 wave32s)
    __shared__ float wdot[8][8];
    __shared__ float wnrm[8][8];
    const int lane = tid & 31, wid = tid >> 5;
    #pragma unroll
    for (int cc = 0; cc < 8; ++cc) {
        float dv = dotv[cc], nv = nrmv[cc];
        for (int off = 16; off > 0; off >>= 1) {
            dv += __shfl_down(dv, off, 32);
            nv += __shfl_down(nv, off, 32);
        }
        if (lane == 0) { wdot[cc][wid] = dv; wnrm[cc][wid] = nv; }
    }
    __syncthreads();
    if (tid < 8) {
        float dv = 0.0f, nv = 0.0f;
        #pragma unroll
        for (int w = 0; w < 8; ++w) { dv += wdot[tid][w]; nv += wnrm[tid][w]; }
        out[i * Bsz + (c0 + tid)] = dv / (sqrtf(nv) + 1e-8f);
    }
}

// ---------------------------------------------------------------------------
extern "C" void kernel_launch(void* const* d_in, const int* in_sizes, int n_in,
                              void* d_out, int out_size, void* d_ws, size_t ws_size,
                              hipStream_t stream) {
    (void)in_sizes; (void)n_in; (void)out_size; (void)ws_size;

    const float* img      = (const float*)d_in[0];   // (48,36,1024)
    const float* cap      = (const float*)d_in[1];   // (48,50,1024)
    const int*   lens     = (const int*)  d_in[2];   // (48,)
    const float* W_gamma  = (const float*)d_in[3];   // (1024,1024)
    const float* b_gamma  = (const float*)d_in[4];   // (1024,)
    const float* W_beta   = (const float*)d_in[5];   // (1024,1024)
    const float* b_beta   = (const float*)d_in[6];   // (1024,)
    float*       out      = (float*)d_out;           // (48,48)

    float* ws = (float*)d_ws;
    float* mean_     = ws;                         // 1024
    float* rsig_     = mean_    + Dd;              // 1024
    float* cap_repr_ = rsig_    + Dd;              // 48*1024
    float* capn_     = cap_repr_ + Bsz * Dd;       // 48*1024
    float* gammas_   = capn_    + Bsz * Dd;        // 48*1024
    float* betas_    = gammas_  + Bsz * Dd;        // 48*1024   (~795 KB total)

    // 1) BN stats (independent)
    bn_stats_kernel<<<dim3(Dd / 256), dim3(256), 0, stream>>>(img, mean_, rsig_);
    // 2) caption pooling + fused L2 norm (independent)
    cap_pool_kernel<<<dim3(Bsz), dim3(256), 0, stream>>>(cap, lens, cap_repr_, capn_);
    // 3) FiLM GEMMs (fp32 WMMA): grid = (N tiles, M tiles, {gamma,beta})
    film_gemm_kernel<<<dim3(Dd / 16, Bsz / 16, 2), dim3(32), 0, stream>>>(
        cap_repr_, W_gamma, b_gamma, W_beta, b_beta, gammas_, betas_);
    // 4) fused fovea + cosine similarities: grid = (images, caption tiles of 8)
    fovea_sims_kernel<<<dim3(Bsz, Bsz / 8), dim3(256), 0, stream>>>(
        img, mean_, rsig_, gammas_, betas_, capn_, out);
}